// UpS_30442728194063
// MI455X (gfx1250) — compile-verified
//
#include <hip/hip_runtime.h>
#include <stdint.h>

typedef float v2f __attribute__((ext_vector_type(2)));
typedef float v8f __attribute__((ext_vector_type(8)));

#define N_NODES 65536
#define NGRAPH  64
#define NPG     1024
#define KDIM    128
#define DIN     128
#define DOUT    128
#define NEDGE   1048576
#define EPG     (NEDGE / NGRAPH)     // 16384 edges per graph (contiguous in edge list)
#define CHUNK   128                  // dst nodes per scatter block
#define NCHUNK  (NPG / CHUNK)        // 8 chunks per graph

// ---- fused-GEMM LDS plan (floats):
//  phase 1: WselfT [128 cols][34] at 0 (4352), WneighT [128][34] at 4352  (8704 total)
//  phase 2: assign tiles 4 waves x [16][132] at 0 (8448), feaT [128][34] at 8448 (4352)
#define LDS_FLOATS 12800
#define WN_OFF  4352
#define FEA_OFF 8448
#define TPAD    34     // transposed pad (even -> 8B-aligned b64 frag loads, conflict-free)
#define APAD    132    // assign row pad

__global__ void US_zero_deg(float* __restrict__ deg) {
    int i = blockIdx.x * blockDim.x + threadIdx.x;
    ((float4*)deg)[i] = make_float4(0.f, 0.f, 0.f, 0.f);
}

// Privatized scatter: one block per (graph, 128-node dst chunk).
// LDS accumulator (64KB) takes all feature adds; global writeback is plain stores.
__global__ __launch_bounds__(256)
void US_scatter_lds(const float* __restrict__ h,
                    const int* __restrict__ esrc,
                    const int* __restrict__ edst,
                    float* __restrict__ deg,
                    float* __restrict__ neigh) {
    __shared__ __align__(16) float acc[CHUNK * DIN];   // 65536 bytes
    const int tid   = threadIdx.x;
    const int wave  = tid >> 5;                        // 0..7
    const int lane  = tid & 31;
    const int g     = blockIdx.x >> 3;                 // graph id
    const int chunk = blockIdx.x & (NCHUNK - 1);
    const int nodeBase = g * NPG + chunk * CHUNK;      // global first node of chunk
    const int ebase = g * EPG;

    // zero the LDS accumulator
#pragma unroll
    for (int i = 0; i < (CHUNK * DIN / 4) / 256; ++i)
        ((float4*)acc)[i * 256 + tid] = make_float4(0.f, 0.f, 0.f, 0.f);
    __syncthreads();

    // each wave scans a contiguous 1/8 of this graph's edges, 32 at a time
    const int estart = ebase + wave * (EPG / 8);
    for (int base = 0; base < EPG / 8; base += 32) {
        const int e = estart + base + lane;
        const int d32 = edst[e];
        const int s32 = esrc[e];
        if (chunk == 0) atomicAdd(deg + d32, 1.0f);    // degree: once per edge overall
#pragma unroll 4
        for (int j = 0; j < 32; ++j) {
            const int d = __shfl(d32, j);
            const int local = d - nodeBase;
            if ((unsigned)local < (unsigned)CHUNK) {   // wave-uniform branch
                const int s = __shfl(s32, j);
                const float4 v = ((const float4*)(h + (size_t)s * DIN))[lane];
                float* a = acc + local * DIN + lane * 4;
                atomicAdd(a + 0, v.x);
                atomicAdd(a + 1, v.y);
                atomicAdd(a + 2, v.z);
                atomicAdd(a + 3, v.w);
            }
        }
    }
    __syncthreads();

    // writeback (each node owned by exactly one block -> plain stores)
    float4* dst = (float4*)(neigh + (size_t)nodeBase * DIN);
#pragma unroll
    for (int i = 0; i < (CHUNK * DIN / 4) / 256; ++i)
        dst[i * 256 + tid] = ((float4*)acc)[i * 256 + tid];
}

__global__ __launch_bounds__(128)
void US_fused_gemm(const float* __restrict__ h,
                   const float* __restrict__ fea,
                   const float* __restrict__ Wself,
                   const float* __restrict__ Wneigh,
                   const float* __restrict__ bias,
                   const float* __restrict__ deg,
                   const float* __restrict__ neigh,
                   float* __restrict__ out) {
    __shared__ __align__(16) float lds[LDS_FLOATS];
    const int tid  = threadIdx.x;
    const int wave = tid >> 5;
    const int lane = tid & 31;
    const int half = lane >> 4;          // 0 or 1
    const int lm   = lane & 15;
    const int rowBase = blockIdx.x * 64 + wave * 16;  // this wave's 16 rows
    const int g = (blockIdx.x * 64) / NPG;            // graph id (block within one graph)

    const int arow = rowBase + lm;
    const float invd = 1.0f / fmaxf(deg[arow], 1.0f);

    v8f C[8];
#pragma unroll
    for (int t = 0; t < 8; ++t) C[t] = (v8f){0.f,0.f,0.f,0.f,0.f,0.f,0.f,0.f};

    // ---------- Phase 1: S = h@Wself + (neigh*invd)@Wneigh ----------
    for (int kc = 0; kc < 4; ++kc) {
        __syncthreads();
        // stage 32 K-rows of both weight matrices TRANSPOSED: ldsT[col*TPAD + krow]
        {
            const float4* ws4 = (const float4*)(Wself  + (size_t)kc * 32 * KDIM);
            const float4* wn4 = (const float4*)(Wneigh + (size_t)kc * 32 * KDIM);
#pragma unroll
            for (int i = 0; i < 8; ++i) {
                int idx = i * 128 + tid;       // 0..1023 float4 slots
                int r   = idx >> 5;            // K-row 0..31
                int cb  = (idx & 31) * 4;      // first column of the float4
                float4 ws = ws4[idx];
                float4 wn = wn4[idx];
                lds[(cb + 0) * TPAD + r] = ws.x;
                lds[(cb + 1) * TPAD + r] = ws.y;
                lds[(cb + 2) * TPAD + r] = ws.z;
                lds[(cb + 3) * TPAD + r] = ws.w;
                lds[WN_OFF + (cb + 0) * TPAD + r] = wn.x;
                lds[WN_OFF + (cb + 1) * TPAD + r] = wn.y;
                lds[WN_OFF + (cb + 2) * TPAD + r] = wn.z;
                lds[WN_OFF + (cb + 3) * TPAD + r] = wn.w;
            }
        }
        __syncthreads();
#pragma unroll
        for (int k = 0; k < 32; k += 4) {
            const int kk = kc * 32 + k + half * 2;
            v2f a_s = *(const v2f*)(h     + (size_t)arow * DIN + kk);
            v2f a_n = *(const v2f*)(neigh + (size_t)arow * DIN + kk);
            a_n *= invd;   // fold mean-normalization into A fragment
#pragma unroll
            for (int t = 0; t < 8; ++t) {
                const int col = t * 16 + lm;
                v2f b_s = *(const v2f*)(&lds[col * TPAD + k + half * 2]);
                C[t] = __builtin_amdgcn_wmma_f32_16x16x4_f32(
                           false, a_s, false, b_s, (short)0, C[t], false, false);
                v2f b_n = *(const v2f*)(&lds[WN_OFF + col * TPAD + k + half * 2]);
                C[t] = __builtin_amdgcn_wmma_f32_16x16x4_f32(
                           false, a_n, false, b_n, (short)0, C[t], false, false);
            }
        }
    }

    // ---------- bias + relu ----------
#pragma unroll
    for (int t = 0; t < 8; ++t) {
        float bt = bias[t * 16 + lm];
#pragma unroll
        for (int v = 0; v < 8; ++v) C[t][v] = fmaxf(C[t][v] + bt, 0.0f);
    }

    // ---------- softmax over 128 cols per row (row = v + 8*half) ----------
#pragma unroll
    for (int v = 0; v < 8; ++v) {
        float m = C[0][v];
#pragma unroll
        for (int t = 1; t < 8; ++t) m = fmaxf(m, C[t][v]);
#pragma unroll
        for (int off = 1; off < 16; off <<= 1) m = fmaxf(m, __shfl_xor(m, off, 32));
        float e[8];
        float s = 0.f;
#pragma unroll
        for (int t = 0; t < 8; ++t) { e[t] = __expf(C[t][v] - m); s += e[t]; }
#pragma unroll
        for (int off = 1; off < 16; off <<= 1) s += __shfl_xor(s, off, 32);
        const float inv = 1.0f / s;
#pragma unroll
        for (int t = 0; t < 8; ++t) C[t][v] = e[t] * inv;
    }

    __syncthreads();  // all waves done reading W region before overwrite

    // ---------- stage assign tile (16x128, pad 132) per wave ----------
    float* al = lds + wave * 16 * APAD;
#pragma unroll
    for (int t = 0; t < 8; ++t)
#pragma unroll
        for (int v = 0; v < 8; ++v)
            al[(v + 8 * half) * APAD + t * 16 + lm] = C[t][v];

#pragma unroll
    for (int t = 0; t < 8; ++t) C[t] = (v8f){0.f,0.f,0.f,0.f,0.f,0.f,0.f,0.f};

    // ---------- Phase 2: out = assign @ fea[g] ----------
    const float* feag = fea + (size_t)g * KDIM * DOUT;
    float* fl = lds + FEA_OFF;
    for (int kc = 0; kc < 4; ++kc) {
        __syncthreads();
        const float4* f4 = (const float4*)(feag + (size_t)kc * 32 * DOUT);
#pragma unroll
        for (int i = 0; i < 8; ++i) {
            int idx = i * 128 + tid;
            int r   = idx >> 5;
            int cb  = (idx & 31) * 4;
            float4 fv = f4[idx];
            fl[(cb + 0) * TPAD + r] = fv.x;
            fl[(cb + 1) * TPAD + r] = fv.y;
            fl[(cb + 2) * TPAD + r] = fv.z;
            fl[(cb + 3) * TPAD + r] = fv.w;
        }
        __syncthreads();
#pragma unroll
        for (int k = 0; k < 32; k += 4) {
            const int kk = kc * 32 + k + half * 2;
            v2f a = *(const v2f*)(&al[lm * APAD + kk]);
#pragma unroll
            for (int t = 0; t < 8; ++t) {
                const int col = t * 16 + lm;
                v2f b = *(const v2f*)(&fl[col * TPAD + k + half * 2]);
                C[t] = __builtin_amdgcn_wmma_f32_16x16x4_f32(
                           false, a, false, b, (short)0, C[t], false, false);
            }
        }
    }

    // ---------- store ----------
#pragma unroll
    for (int t = 0; t < 8; ++t)
#pragma unroll
        for (int v = 0; v < 8; ++v)
            out[(size_t)(rowBase + v + 8 * half) * DOUT + t * 16 + lm] = C[t][v];
}

extern "C" void kernel_launch(void* const* d_in, const int* in_sizes, int n_in,
                              void* d_out, int out_size, void* d_ws, size_t ws_size,
                              hipStream_t stream) {
    (void)in_sizes; (void)n_in; (void)out_size; (void)ws_size;
    const float* h      = (const float*)d_in[0];
    const float* fea    = (const float*)d_in[1];
    const float* Wself  = (const float*)d_in[2];
    const float* Wneigh = (const float*)d_in[3];
    const float* bias   = (const float*)d_in[4];
    const int*   esrc   = (const int*)d_in[5];
    const int*   edst   = (const int*)d_in[6];
    float* out = (float*)d_out;

    float* deg   = (float*)d_ws;            // [N]
    float* neigh = deg + N_NODES;           // [N*128]

    // zero deg only (neigh is fully overwritten by the scatter writeback)
    US_zero_deg<<<(N_NODES / 4) / 256, 256, 0, stream>>>(deg);

    // privatized scatter: 64 graphs x 8 dst-chunks, LDS accumulation
    US_scatter_lds<<<NGRAPH * NCHUNK, 256, 0, stream>>>(h, esrc, edst, deg, neigh);

    // fused GEMM1 + bias + relu + softmax + per-graph GEMM2
    US_fused_gemm<<<N_NODES / 64, 128, 0, stream>>>(h, fea, Wself, Wneigh, bias,
                                                    deg, neigh, out);
}